// VectorQuantize_27685359190171
// MI455X (gfx1250) — compile-verified
//
#include <hip/hip_runtime.h>
#include <hip/hip_bf16.h>

typedef __attribute__((ext_vector_type(2))) float v2f;
typedef __attribute__((ext_vector_type(8))) float v8f;

#define CODEBOOK 8192
#define DIM      64
#define NROWS    (4 * 4096)       // B*N = 16384
#define WPB      8                // waves per block (wave32)

// ---------------------------------------------------------------------------
// Kernel 0: per-code squared norms + zero the counts histogram (ws scratch)
// ---------------------------------------------------------------------------
__global__ void vq_prep_kernel(const float* __restrict__ embed,
                               float* __restrict__ enorm,
                               float* __restrict__ counts) {
  int k = blockIdx.x * blockDim.x + threadIdx.x;
  if (k < CODEBOOK) {
    const float4* e = (const float4*)(embed + (size_t)k * DIM);
    float s = 0.f;
#pragma unroll
    for (int i = 0; i < DIM / 4; ++i) {
      float4 v = e[i];
      s += v.x * v.x + v.y * v.y + v.z * v.z + v.w * v.w;
    }
    enorm[k]  = s;
    counts[k] = 0.f;
  }
}

// ---------------------------------------------------------------------------
// Kernel 1: fused  (2*x@E^T - |e|^2)  +  row argmax  +  gather  +  counts
// One wave handles a 16-row tile over the full 8192-code sweep.
// ---------------------------------------------------------------------------
__global__ void __launch_bounds__(32 * WPB)
vq_main_kernel(const float* __restrict__ x,
               const float* __restrict__ node_mask,
               const float* __restrict__ embed,
               const float* __restrict__ enorm,
               float* __restrict__ counts,
               float* __restrict__ out_q,      // [NROWS, DIM]
               float* __restrict__ out_ind) {  // [NROWS]
  const int lane    = threadIdx.x & 31;
  const int wave    = threadIdx.x >> 5;
  const int rowBase = (blockIdx.x * WPB + wave) * 16;
  const int lo      = lane & 15;   // M row (A) / N col (B,C)
  const int hi      = lane >> 4;   // K half select

  // ---- A fragments: 16 steps of 16x4 f32 (ISA 7.12.2 layout) -------------
  // lane L (0-15):  VGPR pair = x[M=L][k0+0], x[M=L][k0+1]
  // lane L (16-31): VGPR pair = x[M=L-16][k0+2], x[M=L-16][k0+3]
  v2f afrag[16];
  {
    const float* xr = x + (size_t)(rowBase + lo) * DIM + hi * 2;
#pragma unroll
    for (int s = 0; s < 16; ++s)
      afrag[s] = *(const v2f*)(xr + s * 4);
  }

  float bestv[8];
  int   besti[8];
#pragma unroll
  for (int j = 0; j < 8; ++j) { bestv[j] = -3.4e38f; besti[j] = 0; }

  // B fragment source: B[k=d][n=code] = embed[code][d]; per step s a lane
  // holds d = s*4 + hi*2 (+1) of code = kt + lo  (mirror of A layout).
  const float* ecol = embed + (size_t)lo * DIM + hi * 2;

  for (int kt = 0; kt < CODEBOOK; kt += 16) {
    const float* eb = ecol + (size_t)kt * DIM;
    // speculative prefetch of next code tile (global_prefetch_b8)
    __builtin_prefetch(eb + 16 * DIM, 0, 1);

    v8f acc = {};
#pragma unroll
    for (int s = 0; s < 16; ++s) {
      v2f b = *(const v2f*)(eb + s * 4);
      acc = __builtin_amdgcn_wmma_f32_16x16x4_f32(
          /*neg_a=*/false, afrag[s], /*neg_b=*/false, b,
          /*c_mod=*/(short)0, acc, /*reuse_a=*/false, /*reuse_b=*/false);
    }

    const float en   = enorm[kt + lo];
    const int   code = kt + lo;
#pragma unroll
    for (int j = 0; j < 8; ++j) {
      float sc = 2.0f * acc[j] - en;            // argmax-equivalent score
      if (sc > bestv[j]) { bestv[j] = sc; besti[j] = code; }
    }
  }

  // ---- argmax across the 16 lanes holding each row (xor stays in half) ---
#pragma unroll
  for (int m = 1; m < 16; m <<= 1) {
#pragma unroll
    for (int j = 0; j < 8; ++j) {
      float ov = __shfl_xor(bestv[j], m, 32);
      int   oi = __shfl_xor(besti[j], m, 32);
      if (ov > bestv[j] || (ov == bestv[j] && oi < besti[j])) {
        bestv[j] = ov; besti[j] = oi;
      }
    }
  }
  // lane 0 now has rows rowBase+0..7 (j=0..7); lane 16 has rows rowBase+8..15.

  __shared__ int s_idx[WPB][16];
  if (lane == 0 || lane == 16) {
    const int rofs = hi * 8;
#pragma unroll
    for (int j = 0; j < 8; ++j) {
      const int r   = rofs + j;
      const int idx = besti[j];
      s_idx[wave][r]       = idx;
      out_ind[rowBase + r] = (float)idx;
      atomicAdd(&counts[idx], node_mask[rowBase + r]);
    }
  }
  __syncthreads();

  // ---- codebook gather: 16 rows x 16 float4 = 256 vec4, 8 per lane -------
#pragma unroll
  for (int t = 0; t < 8; ++t) {
    const int i    = t * 32 + lane;       // 0..255
    const int r    = i >> 4;              // row in tile
    const int c    = i & 15;              // float4 column
    const int code = s_idx[wave][r];
    float4 v = ((const float4*)(embed + (size_t)code * DIM))[c];
    ((float4*)(out_q + (size_t)(rowBase + r) * DIM))[c] = v;
  }
}

// ---------------------------------------------------------------------------
// Kernel 2: perplexity = exp(-sum p*log(p+1e-10)), p = counts/NROWS
// ---------------------------------------------------------------------------
__global__ void vq_perplexity_kernel(const float* __restrict__ counts,
                                     float* __restrict__ out) {
  __shared__ float red[256];
  float s = 0.f;
  for (int k = threadIdx.x; k < CODEBOOK; k += 256) {
    float p = counts[k] * (1.0f / (float)NROWS);
    s += p * __logf(p + 1e-10f);
  }
  red[threadIdx.x] = s;
  __syncthreads();
  for (int o = 128; o > 0; o >>= 1) {
    if ((int)threadIdx.x < o) red[threadIdx.x] += red[threadIdx.x + o];
    __syncthreads();
  }
  if (threadIdx.x == 0) out[0] = __expf(-red[0]);
}

// ---------------------------------------------------------------------------
extern "C" void kernel_launch(void* const* d_in, const int* in_sizes, int n_in,
                              void* d_out, int out_size, void* d_ws, size_t ws_size,
                              hipStream_t stream) {
  const float* x         = (const float*)d_in[0];  // [4,4096,64]
  const float* node_mask = (const float*)d_in[1];  // [4,4096]
  const float* embed     = (const float*)d_in[2];  // [8192,64]

  float* out    = (float*)d_out;
  float* out_q  = out;                         // NROWS*DIM floats
  float* out_i  = out + (size_t)NROWS * DIM;   // NROWS floats
  float* out_p  = out_i + NROWS;               // 1 float

  float* enorm  = (float*)d_ws;                // CODEBOOK floats
  float* counts = enorm + CODEBOOK;            // CODEBOOK floats

  vq_prep_kernel<<<(CODEBOOK + 255) / 256, 256, 0, stream>>>(embed, enorm, counts);

  const int blocks = NROWS / (16 * WPB);       // 128 blocks of 8 waves
  vq_main_kernel<<<blocks, 32 * WPB, 0, stream>>>(x, node_mask, embed, enorm,
                                                  counts, out_q, out_i);

  vq_perplexity_kernel<<<1, 256, 0, stream>>>(counts, out_p);
}